// WriteHead_11948599017548
// MI455X (gfx1250) — compile-verified
//
#include <hip/hip_runtime.h>
#include <math.h>

#define B_ 64
#define D_ 256
#define N_ 2048
#define M_ 512
#define EPS_ 1e-8f

typedef __attribute__((ext_vector_type(2))) float v2f;
typedef __attribute__((ext_vector_type(4))) float v4f;
typedef __attribute__((ext_vector_type(8))) float v8f;

__device__ __forceinline__ float fast_sigmoid(float v) {
    // v_rcp_f32 (~1 ulp) instead of the IEEE v_div_scale/v_div_fixup chain
    return __builtin_amdgcn_rcpf(1.0f + __expf(-v));
}

// ---------------------------------------------------------------------------
// Kernel 1: k = x@Wk+bk, e = sigmoid(x@We+be), a = x@Wa+ba via WMMA f32 16x16x4
// One wave per 16x16 output tile. grid = (M/16, B/16, 3 matrices).
// A 16x4 layout: lanes 0-15 hold K=0(v0),K=1(v1); lanes 16-31 hold K=2,K=3.
// B 4x16 layout: lanes 0-15 = col N, K=0,1; lanes 16-31 = col N, K=2,3.
// C/D: VGPR r -> row r (lanes 0-15) / row r+8 (lanes 16-31), col = lane&15.
// ---------------------------------------------------------------------------
__global__ __launch_bounds__(32)
void proj_wmma(const float* __restrict__ X,
               const float* __restrict__ Wk, const float* __restrict__ bk,
               const float* __restrict__ We, const float* __restrict__ be,
               const float* __restrict__ Wa, const float* __restrict__ ba,
               float* __restrict__ k_out, float* __restrict__ e_out,
               float* __restrict__ a_out)
{
    const int tn  = blockIdx.x;   // output col tile (0..31)
    const int tm  = blockIdx.y;   // output row tile (0..3)
    const int mat = blockIdx.z;   // 0=k, 1=e, 2=a

    const float* W; const float* bias; float* out;
    if (mat == 0)      { W = Wk; bias = bk; out = k_out; }
    else if (mat == 1) { W = We; bias = be; out = e_out; }
    else               { W = Wa; bias = ba; out = a_out; }

    const int lane = threadIdx.x;
    const int half = lane >> 4;      // 0: K 0-1, 1: K 2-3
    const int l15  = lane & 15;
    const int row  = tm * 16 + l15;  // A row
    const int col  = tn * 16 + l15;  // B col

    v8f c = {};
    for (int kk = 0; kk < D_ / 4; ++kk) {
        const int kb = kk * 4 + half * 2;
        v2f af, bf;
        af[0] = X[row * D_ + kb + 0];
        af[1] = X[row * D_ + kb + 1];
        bf[0] = W[(kb + 0) * M_ + col];
        bf[1] = W[(kb + 1) * M_ + col];
        c = __builtin_amdgcn_wmma_f32_16x16x4_f32(
                /*neg_a=*/false, af, /*neg_b=*/false, bf,
                /*c_mod=*/(short)0, c, /*reuse_a=*/false, /*reuse_b=*/false);
    }

    const float bv = bias[col];
    #pragma unroll
    for (int r = 0; r < 8; ++r) {
        const int orow = tm * 16 + r + half * 8;
        float v = c[r] + bv;
        if (mat == 1) v = fast_sigmoid(v);   // sigmoid for e
        out[orow * M_ + col] = v;
    }
}

// ---------------------------------------------------------------------------
// Kernel 2: beta = softplus(x@Wbeta + bbeta), g = sigmoid(x@Wg + bg). grid = B.
// ---------------------------------------------------------------------------
__global__ __launch_bounds__(256)
void betag_kernel(const float* __restrict__ X,
                  const float* __restrict__ Wbeta, const float* __restrict__ bbeta,
                  const float* __restrict__ Wg,    const float* __restrict__ bg,
                  float* __restrict__ beta_out, float* __restrict__ g_out)
{
    __shared__ float rb[256];
    __shared__ float rg[256];
    const int b = blockIdx.x, t = threadIdx.x;
    const float xv = X[b * D_ + t];
    rb[t] = xv * Wbeta[t];
    rg[t] = xv * Wg[t];
    __syncthreads();
    for (int s = 128; s > 0; s >>= 1) {
        if (t < s) { rb[t] += rb[t + s]; rg[t] += rg[t + s]; }
        __syncthreads();
    }
    if (t == 0) {
        const float zb = rb[0] + bbeta[0];
        beta_out[b] = (zb > 20.0f) ? zb : log1pf(__expf(zb));   // softplus
        g_out[b] = fast_sigmoid(rg[0] + bg[0]);
    }
}

// ---------------------------------------------------------------------------
// Kernel 3: pass 1 over memory (268 MB, streamed once, non-temporal).
// dot[b,n] = <mem[b,n,:], k[b,:]>, nm = ||row||. One wave per row, B128 loads,
// k staged in LDS. grid = (N/64, B), block = 256 (8 waves x 8 rows).
// ---------------------------------------------------------------------------
__global__ __launch_bounds__(256)
void dotnorm_kernel(const float* __restrict__ mem, const float* __restrict__ k,
                    float* __restrict__ dot_out, float* __restrict__ nm_out)
{
    __shared__ float ks[M_];
    const int b = blockIdx.y;
    const int ntile = blockIdx.x;
    const int t = threadIdx.x;
    for (int i = t; i < M_; i += 256) ks[i] = k[b * M_ + i];
    __syncthreads();

    const int wave = t >> 5, lane = t & 31;
    for (int r = 0; r < 8; ++r) {
        const int n = ntile * 64 + wave * 8 + r;
        const v4f* rowp = (const v4f*)(mem + ((size_t)(b * N_ + n)) * M_);
        float dot = 0.f, sq = 0.f;
        #pragma unroll
        for (int cchunk = 0; cchunk < 4; ++cchunk) {
            const int m4 = cchunk * 32 + lane;
            const v4f v = __builtin_nontemporal_load(rowp + m4);  // th:NT stream
            const int m = m4 * 4;
            dot += v[0] * ks[m] + v[1] * ks[m + 1] + v[2] * ks[m + 2] + v[3] * ks[m + 3];
            sq  += v[0] * v[0] + v[1] * v[1] + v[2] * v[2] + v[3] * v[3];
        }
        #pragma unroll
        for (int off = 16; off >= 1; off >>= 1) {
            dot += __shfl_xor(dot, off, 32);
            sq  += __shfl_xor(sq,  off, 32);
        }
        if (lane == 0) {
            dot_out[b * N_ + n] = dot;
            nm_out[b * N_ + n]  = sqrtf(sq);
        }
    }
}

// ---------------------------------------------------------------------------
// Kernel 4: cos -> softmax over N -> state blend. grid = B, block = 1024.
// ---------------------------------------------------------------------------
__global__ __launch_bounds__(1024)
void state_kernel(const float* __restrict__ k, const float* __restrict__ dot_v,
                  const float* __restrict__ nm_v, const float* __restrict__ beta_v,
                  const float* __restrict__ g_v,  const float* __restrict__ prev,
                  float* __restrict__ state_out)
{
    __shared__ float red[1024];
    const int b = blockIdx.x, t = threadIdx.x;

    // nk = max(||k[b,:]||, eps)
    float pk = 0.f;
    if (t < M_) { const float kv = k[b * M_ + t]; pk = kv * kv; }
    red[t] = pk; __syncthreads();
    for (int s = 512; s > 0; s >>= 1) { if (t < s) red[t] += red[t + s]; __syncthreads(); }
    const float nk = fmaxf(sqrtf(red[0]), EPS_);
    __syncthreads();

    const float beta = beta_v[b], g = g_v[b];
    const int n0 = t, n1 = t + 1024;
    const float c0 = dot_v[b * N_ + n0] / (fmaxf(nm_v[b * N_ + n0], EPS_) * nk);
    const float c1 = dot_v[b * N_ + n1] / (fmaxf(nm_v[b * N_ + n1], EPS_) * nk);
    const float z0 = beta * c0, z1 = beta * c1;

    // max
    red[t] = fmaxf(z0, z1); __syncthreads();
    for (int s = 512; s > 0; s >>= 1) { if (t < s) red[t] = fmaxf(red[t], red[t + s]); __syncthreads(); }
    const float mx = red[0];
    __syncthreads();

    // sum of exp
    const float e0 = __expf(z0 - mx), e1 = __expf(z1 - mx);
    red[t] = e0 + e1; __syncthreads();
    for (int s = 512; s > 0; s >>= 1) { if (t < s) red[t] += red[t + s]; __syncthreads(); }
    const float inv = 1.0f / red[0];

    state_out[b * N_ + n0] = g * (e0 * inv) + (1.f - g) * prev[b * N_ + n0];
    state_out[b * N_ + n1] = g * (e1 * inv) + (1.f - g) * prev[b * N_ + n1];
}

// ---------------------------------------------------------------------------
// Kernel 5: pass 2 over memory (268 MB NT read + 268 MB NT write).
// new_mem = mem*(1 - s*e) + s*a; partial read sums to workspace
// (deterministic two-stage reduction, no float atomics).
// grid = (16, B), block = 256: 128 column-owner lanes (float4 each) x 2 row
// halves of 64 rows. state slice staged in LDS.
// ---------------------------------------------------------------------------
__global__ __launch_bounds__(256)
void update_kernel(const float* __restrict__ mem, const float* __restrict__ state,
                   const float* __restrict__ e, const float* __restrict__ a,
                   float* __restrict__ newmem, float* __restrict__ partial)
{
    __shared__ float ss[128];
    const int b    = blockIdx.y;
    const int tile = blockIdx.x;        // 0..15 (128 rows each)
    const int t    = threadIdx.x;
    const int cidx = t & 127;           // column owner: 4 floats
    const int half = t >> 7;            // row half: 0 or 1
    const int m0   = cidx * 4;
    const int nbase = tile * 128;

    if (t < 128) ss[t] = state[b * N_ + nbase + t];
    __syncthreads();

    const v4f ev = *(const v4f*)(e + b * M_ + m0);
    const v4f av = *(const v4f*)(a + b * M_ + m0);
    v4f acc = {};

    for (int r = 0; r < 64; ++r) {
        const int n = nbase + half * 64 + r;
        const float s = ss[half * 64 + r];
        const size_t idx = ((size_t)(b * N_ + n)) * M_ + m0;
        const v4f mv = __builtin_nontemporal_load((const v4f*)(mem + idx));   // th:NT
        v4f nv;
        #pragma unroll
        for (int j = 0; j < 4; ++j) {
            nv[j]  = mv[j] * (1.f - s * ev[j]) + s * av[j];
            acc[j] += s * mv[j];
        }
        __builtin_nontemporal_store(nv, (v4f*)(newmem + idx));                // th:NT
    }
    // 32 partial slices per b: slice = tile*2 + half
    const size_t pidx = ((size_t)(b * 32 + tile * 2 + half)) * M_ + m0;
    *(v4f*)(partial + pidx) = acc;
}

// ---------------------------------------------------------------------------
// Kernel 6: read[b,m] = sum over 32 partials. grid = B, block = 512.
// ---------------------------------------------------------------------------
__global__ __launch_bounds__(512)
void read_kernel(const float* __restrict__ partial, float* __restrict__ read_out)
{
    const int b = blockIdx.x, m = threadIdx.x;
    float acc = 0.f;
    #pragma unroll
    for (int p = 0; p < 32; ++p)
        acc += partial[((size_t)(b * 32 + p)) * M_ + m];
    read_out[b * M_ + m] = acc;
}

extern "C" void kernel_launch(void* const* d_in, const int* in_sizes, int n_in,
                              void* d_out, int out_size, void* d_ws, size_t ws_size,
                              hipStream_t stream)
{
    const float* x     = (const float*)d_in[0];
    const float* prev  = (const float*)d_in[1];
    const float* mem   = (const float*)d_in[2];
    const float* Wk    = (const float*)d_in[3];
    const float* bk    = (const float*)d_in[4];
    const float* Wbeta = (const float*)d_in[5];
    const float* bbeta = (const float*)d_in[6];
    const float* Wg    = (const float*)d_in[7];
    const float* bg    = (const float*)d_in[8];
    // d_in[9..12] = Ws, bs, Wgamma, bgamma: computed but unused by reference outputs
    const float* We    = (const float*)d_in[13];
    const float* be    = (const float*)d_in[14];
    const float* Wa    = (const float*)d_in[15];
    const float* ba    = (const float*)d_in[16];

    float* out      = (float*)d_out;
    float* read_o   = out;                                  // B*M
    float* state_o  = out + B_ * M_;                        // B*N
    float* newmem_o = out + B_ * M_ + (size_t)B_ * N_;      // B*N*M

    float* ws      = (float*)d_ws;
    float* k_ws    = ws;                                    // B*M
    float* e_ws    = k_ws + B_ * M_;                        // B*M
    float* a_ws    = e_ws + B_ * M_;                        // B*M
    float* beta_ws = a_ws + B_ * M_;                        // B
    float* g_ws    = beta_ws + B_;                          // B
    float* dot_ws  = g_ws + B_;                             // B*N
    float* nm_ws   = dot_ws + (size_t)B_ * N_;              // B*N
    float* part_ws = nm_ws + (size_t)B_ * N_;               // B*32*M

    proj_wmma<<<dim3(M_ / 16, B_ / 16, 3), 32, 0, stream>>>(
        x, Wk, bk, We, be, Wa, ba, k_ws, e_ws, a_ws);
    betag_kernel<<<B_, 256, 0, stream>>>(x, Wbeta, bbeta, Wg, bg, beta_ws, g_ws);
    dotnorm_kernel<<<dim3(N_ / 64, B_), 256, 0, stream>>>(mem, k_ws, dot_ws, nm_ws);
    state_kernel<<<B_, 1024, 0, stream>>>(k_ws, dot_ws, nm_ws, beta_ws, g_ws, prev, state_o);
    update_kernel<<<dim3(16, B_), 256, 0, stream>>>(mem, state_o, e_ws, a_ws, newmem_o, part_ws);
    read_kernel<<<B_, 512, 0, stream>>>(part_ws, read_o);
}